// ExcitaionEmbeddingIonRoPE_32985348833545
// MI455X (gfx1250) — compile-verified
//
#include <hip/hip_runtime.h>

// Problem constants (from the reference): B=256, N=512, D=256, 26 pairs.
#define NPAIRS 26
#define DDIM   256
#define NBATCH 256
#define NSEQ   512
#define ROWS_PER_BLOCK 64
#define THREADS 256
#define ANGLE_SCALE 0.05f

// Native vector types (HIP float4/int2 are structs; builtins need real vectors).
typedef float f4  __attribute__((ext_vector_type(4)));
typedef int   v2i __attribute__((ext_vector_type(2)));

// ---------------------------------------------------------------------------
// Pre-pass: one accurate sincos pair per batch (256 total, negligible).
// angles[b] = (cos(theta), sin(theta), cos(phi), sin(phi))
// ---------------------------------------------------------------------------
__global__ void angles_kernel(const float* __restrict__ n_electrons,
                              const float* __restrict__ n_protons,
                              f4* __restrict__ angles) {
  int b = threadIdx.x;
  if (b < NBATCH) {
    float st, ct, sp, cp;
    sincosf(n_electrons[b] * ANGLE_SCALE, &st, &ct);
    sincosf(n_protons[b]   * ANGLE_SCALE, &sp, &cp);
    f4 v; v.x = ct; v.y = st; v.z = cp; v.w = sp;
    angles[b] = v;
  }
}

// ---------------------------------------------------------------------------
// Main kernel:
//  - async-stage the 26.6 KB embedding table into LDS (CDNA5 ASYNCcnt path)
//  - stage the block's 64 row indices (pair -> LUT -> emb row) into LDS
//  - hot loop: ds_load_b32 (idx) + ds_load_b128 (emb) + rotate + NT b128 store
// Each block owns 64 consecutive rows => exactly one batch (64 | 512), so the
// four rotation coefficients are uniform per block and live in registers.
// ---------------------------------------------------------------------------
__global__ __launch_bounds__(THREADS) void rope_kernel(
    const int* __restrict__ exc,      // [B*N*2] int32 (a,b) pairs
    const float* __restrict__ emb,    // [26*256] f32
    const int* __restrict__ lut,      // [6*6] int32
    const f4* __restrict__ angles,    // [B] (ct, st, cp, sp)
    f4* __restrict__ out)             // [B*N*64] float4 blocks
{
  __shared__ f4  s_emb[NPAIRS * (DDIM / 4)];   // 26624 bytes
  __shared__ int s_idx[ROWS_PER_BLOCK];        // per-row embedding index
  const int tid  = threadIdx.x;
  const int row0 = (int)blockIdx.x * ROWS_PER_BLOCK;

  // ---- Async stage: global -> LDS (tracked by ASYNCcnt) ----
  {
    const unsigned lds_base = (unsigned)(unsigned long long)(void*)&s_emb[0];
    const unsigned long long gbase = (unsigned long long)emb;
    const unsigned total_bytes = NPAIRS * DDIM * 4u;   // 26624
    #pragma unroll 1
    for (unsigned ofs = (unsigned)tid * 16u; ofs < total_bytes; ofs += THREADS * 16u) {
      unsigned lds_addr = lds_base + ofs;
      // GVS mode: global addr = SADDR(64) + VADDR(32); LDS dest addr in VDST.
      asm volatile("global_load_async_to_lds_b128 %0, %1, %2"
                   :
                   : "v"(lds_addr), "v"(ofs), "s"(gbase)
                   : "memory");
    }
  }

  // ---- Stage row metadata: one b64 pair load + LUT lookup per row ----
  if (tid < ROWS_PER_BLOCK) {
    v2i e = ((const v2i*)exc)[row0 + tid];
    s_idx[tid] = lut[e.x * 6 + e.y];
  }

  asm volatile("s_wait_asynccnt 0" ::: "memory");
  __syncthreads();   // orders both the async LDS writes and the ds_stores

  // Uniform per block (one batch per block) -> scalar load, lives in SGPR/VGPR.
  const f4 cs = angles[row0 >> 9];   // (ct, st, cp, sp); NSEQ = 512

  // ---- Rotate + stream out: 64 rows * 64 float4s = 16 items per thread ----
  #pragma unroll 4
  for (int j = 0; j < (ROWS_PER_BLOCK * (DDIM / 4)) / THREADS; ++j) {  // 16 iters
    const int item = j * THREADS + tid;
    const int rloc = item >> 6;           // row within block
    const int g4   = item & 63;           // 4-wide block within the row

    const int idx = s_idx[rloc];                     // ds_load_b32 (broadcast)
    const f4  e   = s_emb[idx * (DDIM / 4) + g4];    // ds_load_b128

    f4 o;
    o.x =  cs.x * e.x + cs.y * e.y;   //  ct*x0 + st*x1
    o.y = -cs.y * e.x + cs.x * e.y;   // -st*x0 + ct*x1
    o.z =  cs.z * e.z + cs.w * e.w;   //  cp*x2 + sp*x3
    o.w = -cs.w * e.z + cs.z * e.w;   // -sp*x2 + cp*x3

    // Write-once 128 MB stream: keep it out of L2 with a nontemporal store.
    __builtin_nontemporal_store(o, &out[(row0 + rloc) * (DDIM / 4) + g4]);
  }
}

// ---------------------------------------------------------------------------
// Harness entry point.
// Inputs (setup_inputs order): excitations[int32 B*N*2], n_electrons[f32 B],
// n_protons[f32 B], emb_weight[f32 26*256], lookup_table[int32 36].
// Output: float32 [B*N*256]. d_ws holds the 256*16B angle table.
// ---------------------------------------------------------------------------
extern "C" void kernel_launch(void* const* d_in, const int* in_sizes, int n_in,
                              void* d_out, int out_size, void* d_ws, size_t ws_size,
                              hipStream_t stream) {
  const int*   exc = (const int*)d_in[0];
  const float* ne  = (const float*)d_in[1];
  const float* np  = (const float*)d_in[2];
  const float* emb = (const float*)d_in[3];
  const int*   lut = (const int*)d_in[4];
  float*       out = (float*)d_out;
  f4*       angles = (f4*)d_ws;

  hipLaunchKernelGGL(angles_kernel, dim3(1), dim3(NBATCH), 0, stream, ne, np, angles);

  const int blocks = (NBATCH * NSEQ) / ROWS_PER_BLOCK;  // 2048
  hipLaunchKernelGGL(rope_kernel, dim3(blocks), dim3(THREADS), 0, stream,
                     exc, emb, lut, angles, (f4*)out);
}